// Detect_33234456937117
// MI455X (gfx1250) — compile-verified
//
#include <hip/hip_runtime.h>
#include <stdint.h>

// ---------------- problem constants (match reference) ----------------
#define N_PRIORS    65536
#define NUM_CLASSES 21
#define CM1         (NUM_CLASSES - 1)          // 20 foreground classes
#define TOTAL       (N_PRIORS * CM1)           // 1,310,720 candidates
#define TOPK        200
#define CONF_T      0.01f
#define NMS_T       0.45f
#define VAR0        0.1f
#define VAR1        0.2f

// ---------------- launch geometry ----------------
#define DEC_THREADS 256
#define DEC_BLOCKS  (N_PRIORS / DEC_THREADS)   // 256
#define SUP_BLOCKS  64
#define SUP_THREADS 256
#define CHUNK       (N_PRIORS / SUP_BLOCKS)    // 1024 priors per block
#define NWAVES      (SUP_THREADS / 32)         // 8 wave32s per block

// ---------------- workspace layout (bytes) ----------------
#define WS_BOXES    0u                               // float4[N_PRIORS]     (1 MB)
#define WS_SCORES   (N_PRIORS * 16u)                 // float[TOTAL]         (5 MB)
#define WS_PART     (WS_SCORES + TOTAL * 4u)         // u64[SUP_BLOCKS]
#define WS_PMAX     (WS_PART + SUP_BLOCKS * 8u)      // float[DEC_BLOCKS]
#define WS_MAXC     (WS_PMAX + DEC_BLOCKS * 4u)      // float[1]
#define WS_NEEDED   (WS_MAXC + 64u)

// Monotone order-preserving float <-> u32 key (handles negatives / -inf).
__device__ __forceinline__ unsigned f2key(float f) {
    unsigned u = __float_as_uint(f);
    return (u & 0x80000000u) ? ~u : (u | 0x80000000u);
}
__device__ __forceinline__ float key2f(unsigned k) {
    unsigned u = (k & 0x80000000u) ? (k & 0x7FFFFFFFu) : ~k;
    return __uint_as_float(u);
}

// =====================================================================
// Kernel 1: SSD box decode + score thresholding + per-block max-coord.
// One thread per prior. Scores stored class-major [c][n] so the NMS
// suppression/argmax slices are coalesced 128-bit streams.
// =====================================================================
__global__ void __launch_bounds__(DEC_THREADS)
decode_init_kernel(const float* __restrict__ loc,
                   const float* __restrict__ conf,
                   const float* __restrict__ prior,
                   float4* __restrict__ boxes,
                   float* __restrict__ scores,
                   float* __restrict__ pmax) {
    const int n = blockIdx.x * DEC_THREADS + threadIdx.x;

    const float l0 = loc[n * 4 + 0], l1 = loc[n * 4 + 1];
    const float l2 = loc[n * 4 + 2], l3 = loc[n * 4 + 3];
    const float p0 = prior[n * 4 + 0], p1 = prior[n * 4 + 1];
    const float p2 = prior[n * 4 + 2], p3 = prior[n * 4 + 3];

    const float cx = fmaf(l0 * VAR0, p2, p0);
    const float cy = fmaf(l1 * VAR0, p3, p1);
    const float hw = p2 * expf(l2 * VAR1) * 0.5f;
    const float hh = p3 * expf(l3 * VAR1) * 0.5f;
    const float x1 = cx - hw, y1 = cy - hh, x2 = cx + hw, y2 = cy + hh;
    boxes[n] = make_float4(x1, y1, x2, y2);

    bool any_valid = false;
    const float* crow = conf + n * NUM_CLASSES + 1;  // skip background
#pragma unroll
    for (int c = 0; c < CM1; ++c) {
        const float s = crow[c];
        const bool v = s > CONF_T;
        any_valid |= v;
        scores[c * N_PRIORS + n] = v ? s : -INFINITY;
    }
    float mc = any_valid ? fmaxf(fmaxf(x1, y1), fmaxf(x2, y2)) : -INFINITY;

    __shared__ float sm[DEC_THREADS];
    sm[threadIdx.x] = mc;
    __syncthreads();
    for (int s = DEC_THREADS / 2; s > 0; s >>= 1) {
        if (threadIdx.x < s) sm[threadIdx.x] = fmaxf(sm[threadIdx.x], sm[threadIdx.x + s]);
        __syncthreads();
    }
    if (threadIdx.x == 0) pmax[blockIdx.x] = sm[0];
}

// =====================================================================
// Kernel 2: reduce per-block max-coords -> single max_coord scalar.
// =====================================================================
__global__ void __launch_bounds__(DEC_BLOCKS)
reduce_maxc_kernel(const float* __restrict__ pmax, float* __restrict__ maxc) {
    __shared__ float sm[DEC_BLOCKS];
    sm[threadIdx.x] = pmax[threadIdx.x];
    __syncthreads();
    for (int s = DEC_BLOCKS / 2; s > 0; s >>= 1) {
        if (threadIdx.x < s) sm[threadIdx.x] = fmaxf(sm[threadIdx.x], sm[threadIdx.x + s]);
        __syncthreads();
    }
    if (threadIdx.x == 0) maxc[0] = sm[0];
}

// =====================================================================
// Kernel 3 (fused NMS iteration, launched 201x):
//   first==1 : only compute per-block argmax partials (no selection).
//   first==0 : (a) all blocks redundantly reduce 64 partials -> winner,
//              (b) block 0 records output row t,
//              (c) each block suppresses its 1024-prior range across all
//                  20 classes (class offsets folded into scalar shift d),
//              (d) same pass produces the next per-block argmax partial.
// Box tile staged into LDS with the CDNA5 async global->LDS copy,
// overlapped with the partials reduction, completed with
// s_wait_asynccnt + workgroup barrier.  Block argmax uses wave32
// shuffle butterflies (barrier-free) + one LDS hop across 8 waves.
// =====================================================================
__global__ void __launch_bounds__(SUP_THREADS)
nms_iter_kernel(const float4* __restrict__ boxes,
                float* __restrict__ scores,
                unsigned long long* __restrict__ partials,
                const float* __restrict__ maxc,
                float* __restrict__ out,
                int t, int first) {
    __shared__ float4 sbox[CHUNK];                   // 16 KB staged box tile
    __shared__ unsigned long long sred[SUP_THREADS]; // select-phase scratch
    __shared__ unsigned long long swave[NWAVES];     // cross-wave argmax hop

    const int tid = threadIdx.x;
    const int blk = blockIdx.x;
    const int n0  = blk * CHUNK;

    // ---- issue async global->LDS staging of this block's box tile ----
#pragma unroll
    for (int j = 0; j < CHUNK / SUP_THREADS; ++j) {
        const int k = tid + j * SUP_THREADS;
        unsigned lds_addr = (unsigned)(uintptr_t)(&sbox[k]);  // low 32b = LDS offset
        unsigned long long gaddr = (unsigned long long)(uintptr_t)(&boxes[n0 + k]);
        asm volatile("global_load_async_to_lds_b128 %0, %1, off"
                     :: "v"(lds_addr), "v"(gaddr) : "memory");
    }

    // ---- select previous winner (uniform branch: 'first' is scalar) ----
    int   sel_idx = -1, sel_cm1 = 0;
    float sel_val = -INFINITY;
    bool  ok = false;
    float unit = 0.0f, sx1 = 0.f, sy1 = 0.f, sx2 = 0.f, sy2 = 0.f, areaS = 0.f;

    if (!first) {
        sred[tid] = (tid < SUP_BLOCKS) ? partials[tid] : 0ull;
        __syncthreads();
        for (int s = SUP_BLOCKS / 2; s > 0; s >>= 1) {  // 64 live entries
            if (tid < s && sred[tid + s] > sred[tid]) sred[tid] = sred[tid + s];
            __syncthreads();
        }
        const unsigned long long best = sred[0];
        sel_idx = (int)(best & 0xFFFFFFFFull);
        sel_val = key2f((unsigned)(best >> 32));
        ok = (sel_val > -INFINITY);                    // NaN compares false too
        sel_cm1 = sel_idx >> 16;                       // N_PRIORS == 1<<16
        const int sel_n = sel_idx & (N_PRIORS - 1);
        unit = maxc[0] + 1.0f;
        const float4 sb = boxes[sel_n];
        sx1 = sb.x; sy1 = sb.y; sx2 = sb.z; sy2 = sb.w;
        areaS = (sx2 - sx1) * (sy2 - sy1);
        if (blk == 0 && tid == 0) {
            float* row = out + (size_t)t * 6;
            if (ok) {
                row[0] = (float)(sel_cm1 + 1);
                row[1] = sel_val;
                row[2] = sx1; row[3] = sy1; row[4] = sx2; row[5] = sy2;
            } else {
                row[0] = 0.f; row[1] = 0.f; row[2] = 0.f;
                row[3] = 0.f; row[4] = 0.f; row[5] = 0.f;
            }
        }
    }

    // ---- complete async staging, make LDS tile visible to all waves ----
    asm volatile("s_wait_asynccnt 0" ::: "memory");
    __syncthreads();

    // ---- fused suppress + next-iteration partial argmax (128-bit IO) ----
    unsigned long long mybest = 0ull;  // key 0 < key(-inf): safe identity
    float4* __restrict__ scores4 = (float4*)scores;
    const int vbase = (n0 >> 2) + tid;           // float4 index of this thread

    for (int c = 0; c < CM1; ++c) {
        const float d = ok ? (float)(c - sel_cm1) * unit : 0.0f;
        if (c + 1 < CM1)
            __builtin_prefetch(&scores4[((c + 1) * N_PRIORS >> 2) + vbase], 0, 1);

        const int vidx = (c * N_PRIORS >> 2) + vbase;     // 16B-aligned
        float4 sv = scores4[vidx];
        float  se[4] = {sv.x, sv.y, sv.z, sv.w};
        const int idx0 = c * N_PRIORS + n0 + tid * 4;
        bool changed = false;

#pragma unroll
        for (int e = 0; e < 4; ++e) {
            float s = se[e];
            const int idx = idx0 + e;
            if (ok && s > -INFINITY) {
                const float4 bx = sbox[tid * 4 + e];
                const float ltx = fmaxf(sx1, bx.x + d);
                const float lty = fmaxf(sy1, bx.y + d);
                const float rbx = fminf(sx2, bx.z + d);
                const float rby = fminf(sy2, bx.w + d);
                const float iw = fmaxf(rbx - ltx, 0.f);
                const float ih = fmaxf(rby - lty, 0.f);
                const float inter = iw * ih;
                const float areaB = (bx.z - bx.x) * (bx.w - bx.y);
                const float iou = inter / (areaS + areaB - inter);
                if (iou > NMS_T || idx == sel_idx) {
                    s = -INFINITY;
                    se[e] = s;
                    changed = true;
                }
            }
            const unsigned long long pk =
                ((unsigned long long)f2key(s) << 32) | (unsigned)idx;
            if (pk > mybest) mybest = pk;
        }
        if (changed)
            scores4[vidx] = make_float4(se[0], se[1], se[2], se[3]);
    }

    // ---- wave32 butterfly argmax (no barriers), then cross-wave hop ----
    unsigned long long v = mybest;
#pragma unroll
    for (int off = 16; off > 0; off >>= 1) {
        const unsigned long long o = __shfl_xor(v, off, 32);
        if (o > v) v = o;
    }
    if ((tid & 31) == 0) swave[tid >> 5] = v;
    __syncthreads();
    if (tid < 32) {
        v = (tid < NWAVES) ? swave[tid] : 0ull;
#pragma unroll
        for (int off = NWAVES / 2; off > 0; off >>= 1) {
            const unsigned long long o = __shfl_xor(v, off, 32);
            if (o > v) v = o;
        }
        if (tid == 0) partials[blk] = v;
    }
}

// =====================================================================
extern "C" void kernel_launch(void* const* d_in, const int* in_sizes, int n_in,
                              void* d_out, int out_size, void* d_ws, size_t ws_size,
                              hipStream_t stream) {
    const float* loc   = (const float*)d_in[0];  // (1, N, 4)
    const float* conf  = (const float*)d_in[1];  // (N, 21)
    const float* prior = (const float*)d_in[2];  // (N, 4)
    float* out = (float*)d_out;                  // (200, 6) f32

    char* ws = (char*)d_ws;
    if (ws_size < WS_NEEDED) return;  // scratch too small; nothing safe to do

    float4*             boxes    = (float4*)(ws + WS_BOXES);
    float*              scores   = (float*)(ws + WS_SCORES);
    unsigned long long* partials = (unsigned long long*)(ws + WS_PART);
    float*              pmax     = (float*)(ws + WS_PMAX);
    float*              maxc     = (float*)(ws + WS_MAXC);

    decode_init_kernel<<<DEC_BLOCKS, DEC_THREADS, 0, stream>>>(
        loc, conf, prior, boxes, scores, pmax);
    reduce_maxc_kernel<<<1, DEC_BLOCKS, 0, stream>>>(pmax, maxc);

    // Priming pass: compute initial per-block argmax partials only.
    nms_iter_kernel<<<SUP_BLOCKS, SUP_THREADS, 0, stream>>>(
        boxes, scores, partials, maxc, out, 0, /*first=*/1);

    // 200 greedy-NMS iterations, each fused select+record+suppress+argmax.
    for (int t = 0; t < TOPK; ++t) {
        nms_iter_kernel<<<SUP_BLOCKS, SUP_THREADS, 0, stream>>>(
            boxes, scores, partials, maxc, out, t, /*first=*/0);
    }
}